// Black_oil_13116830122019
// MI455X (gfx1250) — compile-verified
//
#include <hip/hip_runtime.h>
#include <math.h>

// ---- problem constants -----------------------------------------------------
#define NBb 4
#define NTt 24
#define NZc 22
#define NXc 46
#define NYc 112
#define S_SPATIAL 113344      // 22*46*112
#define SXY       5152        // 46*112
#define NTOT      10881024    // 4*24*113344
#define N4        2720256     // NTOT/4
#define CHUNK     14168       // S_SPATIAL/8
// workspace float offsets
#define WS_PART   0           // [0..799]   partial sums (100 slices x 8 chunks)
#define WS_MEAN   832         // [832..931] pmean[96] then pini_mean[4]
#define WS_SA     960
#define WS_SB     1088
#define WS_SC     1216

typedef float v2f __attribute__((ext_vector_type(2)));
typedef float v8f __attribute__((ext_vector_type(8)));

__device__ __forceinline__ float cleanf(float x) {
  return __builtin_isfinite(x) ? x : 0.0f;
}

// ---- stage 1: WMMA-based spatial sums --------------------------------------
// blocks 0..767 : pressure slice bt = blk>>3, chunk = blk&7
// blocks 768..799: Pini slice b = (blk-768)>>3, chunk = blk&7
__global__ void __launch_bounds__(256) stage1_sum(const float* __restrict__ pressure,
                                                  const float* __restrict__ Pini,
                                                  float* __restrict__ ws) {
  const int blk   = blockIdx.x;
  const int chunk = blk & 7;
  const float* src;
  if (blk < 768) {
    const int bt = blk >> 3;
    src = pressure + (size_t)bt * S_SPATIAL + chunk * CHUNK;
  } else {
    const int b = (blk - 768) >> 3;
    src = Pini + (size_t)b * S_SPATIAL + chunk * CHUNK;
  }
  const int lane = threadIdx.x & 31;
  const int wave = threadIdx.x >> 5;

  const v2f ones = {1.0f, 1.0f};
  v8f acc = {0.f, 0.f, 0.f, 0.f, 0.f, 0.f, 0.f, 0.f};

  // 8 waves x 64 elems = 512 per iter; 28*512 = 14336 >= 14168
  for (int it = 0; it < 28; ++it) {
    const int local = it * 512 + wave * 64 + lane * 2;
    v2f a;
    if (local < CHUNK) a = *(const v2f*)(src + local);
    else               a = (v2f){0.0f, 0.0f};
    // D = A x ones + C : row-sums of 16x4 f32 tile, f32 accumulate
    acc = __builtin_amdgcn_wmma_f32_16x16x4_f32(false, a, false, ones,
                                                (short)0, acc, false, false);
  }

  float s = acc[0] + acc[1] + acc[2] + acc[3] + acc[4] + acc[5] + acc[6] + acc[7];
#pragma unroll
  for (int off = 16; off > 0; off >>= 1) s += __shfl_xor(s, off, 32);

  __shared__ float wsum[8];
  if (lane == 0) wsum[wave] = s;
  __syncthreads();
  if (threadIdx.x == 0) {
    float tot = 0.f;
#pragma unroll
    for (int w = 0; w < 8; ++w) tot += wsum[w];
    ws[WS_PART + blk] = tot * (1.0f / 16.0f);  // undo 16-column replication
  }
}

// ---- stage 2: per-(b,t) scalar coefficients --------------------------------
__global__ void stage2_scalars(float* __restrict__ ws) {
  const int tid = threadIdx.x;
  if (tid < 100) {
    float sum = 0.f;
#pragma unroll
    for (int c = 0; c < 8; ++c) sum += ws[WS_PART + tid * 8 + c];
    ws[WS_MEAN + tid] = sum * (1000.0f / (float)S_SPATIAL);  // mean * MAXP
  }
  __syncthreads();
  if (tid < 96) {
    const int b = tid / 24, t = tid - b * 24;
    const float ap = (t == 0) ? ws[WS_MEAN + 96 + b] : ws[WS_MEAN + tid - 1];
    const float pm = ws[WS_MEAN + tid];

    // mobilities' scalar parts from avg prior pressure
    float UG = 3e-10f * ap * ap + 1e-6f * ap + 0.0133f;
    float BG = expf(-0.0017f * (ap - 14.7f));
    const float e1 = (ap < 203.0f) ? (-8e-5f * (14.7f - ap)) : (-8e-5f * (14.7f - 203.0f));
    const float e2 = (ap < 203.0f) ? 0.0f : (-1e-5f * (ap - 203.0f));
    float BO = expf(-(e1 + e2));
    UG = cleanf(UG); BG = cleanf(BG); BO = cleanf(BO);

    // Peaceman well coefficient from current pressure mean
    const float e1w = (pm < 203.0f) ? (-8e-5f * (14.7f - pm)) : (-8e-5f * (14.7f - 203.0f));
    const float e2w = (pm < 203.0f) ? 0.0f : (-1e-5f * (pm - 203.0f));
    const float bow = expf(-(e1w + e2w));
    const float drawdown = pm - 100.0f;
    const float right = 0.81093021621632877f;  // log(450/200) + SKIN
    const float sC = -fabsf(drawdown) * (2.0f * 3.14159265358979323846f * 100.0f)
                     / (2.5f * bow * right);

    ws[WS_SA + tid] = 1.0f / (2.5f * BO);
    ws[WS_SB + tid] = 1.0f / (UG * BG);
    ws[WS_SC + tid] = sC;
  }
}

// ---- stage 3: streaming elementwise pass (float4) --------------------------
__global__ void __launch_bounds__(256) stage3_elem(const float* __restrict__ water,
                                                   const float* __restrict__ gas,
                                                   const float* __restrict__ timev,
                                                   const float* __restrict__ perm,
                                                   const float* __restrict__ Swini,
                                                   const float* __restrict__ ws,
                                                   float* __restrict__ out) {
  const int tid = blockIdx.x * 256 + threadIdx.x;   // grid sized exactly to N4
  const int i   = tid * 4;
  const int bt  = i / S_SPATIAL;
  const int r   = i - bt * S_SPATIAL;
  const int t   = bt % 24;
  const int b   = bt / 24;
  const int xy  = r % SXY;
  const int x   = xy / NYc;
  const int y0  = xy - x * NYc;       // multiple of 4; quad stays in one row

  const float4 ws4 = *(const float4*)(water + i);
  const float4 gs4 = *(const float4*)(gas + i);
  const float4 tm4 = *(const float4*)(timev + i);
  float4 pw4, pg4, pt4;
  if (t > 0) {
    pw4 = *(const float4*)(water + (i - S_SPATIAL));
    pg4 = *(const float4*)(gas + (i - S_SPATIAL));
    pt4 = *(const float4*)(timev + (i - S_SPATIAL));
  } else {
    const float sini = Swini[0];
    pw4 = make_float4(sini, sini, sini, sini);
    pg4 = make_float4(0.f, 0.f, 0.f, 0.f);
    pt4 = make_float4(0.f, 0.f, 0.f, 0.f);
  }

  const float sA = ws[WS_SA + bt];
  const float sB = ws[WS_SB + bt];
  const float sC = ws[WS_SC + bt];

  const float wsv[4] = {ws4.x, ws4.y, ws4.z, ws4.w};
  const float gsv[4] = {gs4.x, gs4.y, gs4.z, gs4.w};
  const float tmv[4] = {tm4.x, tm4.y, tm4.z, tm4.w};
  const float pwv[4] = {pw4.x, pw4.y, pw4.z, pw4.w};
  const float pgv[4] = {pg4.x, pg4.y, pg4.z, pg4.w};
  const float ptv[4] = {pt4.x, pt4.y, pt4.z, pt4.w};

  const float invden = 1.0f / 0.7f;
  float Mw[4], Mo[4], Mg[4], ds[4], dt[4];
#pragma unroll
  for (int c = 0; c < 4; ++c) {
    ds[c] = wsv[c] - pwv[c];
    dt[c] = cleanf(tmv[c] - ptv[c]);
    const float swn = (pwv[c] - 0.1f) * invden;
    const float gn  = pgv[c] * invden;
    const float a1  = 1.0f - swn;
    const float a2  = 1.0f - gn;
    const float krw = 0.3f * swn * swn;
    const float kro = 0.9f * a1 * a1 * a2 * a2;
    const float krg = 0.8f * gn * gn;
    Mw[c] = cleanf(krw);
    Mo[c] = cleanf(kro * sA);
    Mg[c] = cleanf(krg * sB);
  }

  // qoil: nonzero only at 20 well columns
  float qo[4] = {0.f, 0.f, 0.f, 0.f};
  const unsigned long long WELLX =
      (1ull << 6) | (1ull << 8) | (1ull << 9) | (1ull << 10) | (1ull << 11) |
      (1ull << 12) | (1ull << 13) | (1ull << 14) | (1ull << 15) | (1ull << 17) |
      (1ull << 18) | (1ull << 21) | (1ull << 24) | (1ull << 37);
  if ((WELLX >> x) & 1ull) {
    const int codes[20] = {1598, 1039, 1489, 932, 941, 1036, 1031, 2373, 1483, 2053,
                           2069, 1745, 2724, 1303, 1971, 1410, 4241, 735, 1643, 1147};
    const float4 pm4 = *(const float4*)(perm + (size_t)b * S_SPATIAL + r);
    const float pmv[4] = {pm4.x, pm4.y, pm4.z, pm4.w};
    const int code0 = x * NYc + y0;
#pragma unroll
    for (int c = 0; c < 4; ++c) {
      const int code = code0 + c;
      bool hit = false;
#pragma unroll
      for (int k = 0; k < 20; ++k) hit = hit || (code == codes[k]);
      if (hit) {
        const float swc = (wsv[c] - 0.1f) * invden;
        const float gnc = gsv[c] * invden;
        const float b1 = 1.0f - swc, b2 = 1.0f - gnc;
        const float kroW = 0.9f * b1 * b1 * b2 * b2;
        qo[c] = sC * (pmv[c] * 1000.0f) * kroW;  // = -|drawdown|*J
      }
    }
  }

  float* o = out + i;
  *(float4*)(o + 0 * (size_t)NTOT) = make_float4(qo[0], qo[1], qo[2], qo[3]);
  *(float4*)(o + 1 * (size_t)NTOT) = make_float4(Mw[0], Mw[1], Mw[2], Mw[3]);
  *(float4*)(o + 2 * (size_t)NTOT) = make_float4(Mo[0], Mo[1], Mo[2], Mo[3]);
  *(float4*)(o + 3 * (size_t)NTOT) = make_float4(Mg[0], Mg[1], Mg[2], Mg[3]);
  *(float4*)(o + 4 * (size_t)NTOT) = make_float4(ds[0], ds[1], ds[2], ds[3]);
  *(float4*)(o + 5 * (size_t)NTOT) = make_float4(dt[0], dt[1], dt[2], dt[3]);
}

// ---- launcher --------------------------------------------------------------
extern "C" void kernel_launch(void* const* d_in, const int* in_sizes, int n_in,
                              void* d_out, int out_size, void* d_ws, size_t ws_size,
                              hipStream_t stream) {
  const float* pressure = (const float*)d_in[0];
  const float* perm     = (const float*)d_in[1];
  const float* Pini     = (const float*)d_in[2];
  const float* Swini    = (const float*)d_in[3];
  const float* water    = (const float*)d_in[4];
  const float* gas      = (const float*)d_in[5];
  const float* timev    = (const float*)d_in[6];
  float* out = (float*)d_out;
  float* ws  = (float*)d_ws;

  stage1_sum<<<dim3(800), dim3(256), 0, stream>>>(pressure, Pini, ws);
  stage2_scalars<<<dim3(1), dim3(128), 0, stream>>>(ws);
  stage3_elem<<<dim3(N4 / 256), dim3(256), 0, stream>>>(water, gas, timev, perm,
                                                        Swini, ws, out);
}